// WinnerTakeAllAttention_81003083202667
// MI455X (gfx1250) — compile-verified
//
#include <hip/hip_runtime.h>
#include <math.h>

#define B_ 16
#define N_ 4096
#define C_ 1024
#define K_ 64

typedef __attribute__((ext_vector_type(2))) float v2f;
typedef __attribute__((ext_vector_type(8))) float v8f;

// ---------------------------------------------------------------------------
// Kernel 1: w_mean[c] = mean_d W[d,c];  b_mean = mean(b)
// ---------------------------------------------------------------------------
__global__ __launch_bounds__(256)
void wta_wmean_kernel(const float* __restrict__ W, const float* __restrict__ b,
                      float* __restrict__ w_mean, float* __restrict__ b_mean) {
  const int c = blockIdx.x * 256 + threadIdx.x;   // 0..1023
  float s = 0.0f;
  for (int d = 0; d < C_; ++d) s += W[d * C_ + c];
  w_mean[c] = s * (1.0f / (float)C_);

  if (blockIdx.x == 0) {
    __shared__ float red[256];
    float bs = 0.0f;
    for (int i = threadIdx.x; i < C_; i += 256) bs += b[i];
    red[threadIdx.x] = bs;
    __syncthreads();
    for (int off = 128; off > 0; off >>= 1) {
      if (threadIdx.x < off) red[threadIdx.x] += red[threadIdx.x + off];
      __syncthreads();
    }
    if (threadIdx.x == 0) *b_mean = red[0] * (1.0f / (float)C_);
  }
}

// ---------------------------------------------------------------------------
// Kernel 2: scores[row] = x[row,:].w_mean + b_mean via V_WMMA_F32_16X16X4_F32.
// One wave handles 16 rows. A = 16x4 f32 tile of x (ISA layout: lanes 0-15
// hold K=0,1; lanes 16-31 hold K=2,3). B = 4x16 with w_mean chunk in column
// N=0 only (lane 0 carries K=0,1; lane 16 carries K=2,3). D column 0 = scores.
// ---------------------------------------------------------------------------
__global__ __launch_bounds__(256)
void wta_scores_kernel(const float* __restrict__ x,
                       const float* __restrict__ w_mean,
                       const float* __restrict__ b_mean,
                       float* __restrict__ scores) {
  __shared__ float sw[C_];
  for (int i = threadIdx.x; i < C_; i += 256) sw[i] = w_mean[i];
  __syncthreads();

  const int wave = threadIdx.x >> 5;                 // 0..7
  const int lane = threadIdx.x & 31;
  const int kh   = lane >> 4;                        // 0 or 1 (K half)
  const int r    = lane & 15;                        // row within tile
  const long tile = (long)blockIdx.x * 8 + wave;     // 0..4095
  const long row0 = tile * 16;                       // base row in (B*N)

  const float* xrow = x + (row0 + r) * (long)C_ + 2 * kh;

  v8f acc = {};
#pragma unroll 4
  for (int k = 0; k < C_; k += 4) {
    v2f a = *(const v2f*)(xrow + k);                 // global_load_b64
    v2f wv = *(const v2f*)(&sw[k + 2 * kh]);         // ds_load_b64
    v2f bv;
    bv.x = (r == 0) ? wv.x : 0.0f;                   // cndmask, EXEC stays full
    bv.y = (r == 0) ? wv.y : 0.0f;
    acc = __builtin_amdgcn_wmma_f32_16x16x4_f32(
        /*neg_a=*/false, a, /*neg_b=*/false, bv,
        /*c_mod=*/(short)0, acc, /*reuse_a=*/false, /*reuse_b=*/false);
  }

  if (r == 0) {
    // D column 0: lane 0 -> M=0..7, lane 16 -> M=8..15 (VGPRs 0..7)
    const float bm = *b_mean;
#pragma unroll
    for (int i = 0; i < 8; ++i)
      scores[row0 + kh * 8 + i] = acc[i] + bm;
  }
}

// ---------------------------------------------------------------------------
// Kernel 3: per batch: softmax stats + top-64 extraction (stable tie-break),
// write mask, compacted indices and softmax weights.
// ---------------------------------------------------------------------------
__device__ __forceinline__ void argmax_comb(float& v, int& idx, float v2, int i2) {
  if (v2 > v || (v2 == v && i2 < idx)) { v = v2; idx = i2; }
}

__global__ __launch_bounds__(1024)
void wta_topk_kernel(const float* __restrict__ scores,
                     float* __restrict__ mask_out,
                     float* __restrict__ topk_w,
                     int* __restrict__ topk_idx) {
  const int b    = blockIdx.x;
  const int tid  = threadIdx.x;
  const int lane = tid & 31;
  const int wave = tid >> 5;                         // 0..31

  __shared__ float sred[32];
  __shared__ float wv[32];
  __shared__ int   wi[32];
  __shared__ float sbv;
  __shared__ int   sbi;

  // 4 scores per thread, register-resident
  const float* sb = scores + (long)b * N_;
  float lv[4]; int li[4];
#pragma unroll
  for (int t = 0; t < 4; ++t) { li[t] = tid * 4 + t; lv[t] = sb[tid * 4 + t]; }

  // ---- global max ----
  float m = fmaxf(fmaxf(lv[0], lv[1]), fmaxf(lv[2], lv[3]));
  for (int off = 16; off > 0; off >>= 1) m = fmaxf(m, __shfl_xor(m, off, 32));
  if (lane == 0) sred[wave] = m;
  __syncthreads();
  if (wave == 0) {
    float t = sred[lane];
    for (int off = 16; off > 0; off >>= 1) t = fmaxf(t, __shfl_xor(t, off, 32));
    if (lane == 0) sred[0] = t;
  }
  __syncthreads();
  m = sred[0];
  __syncthreads();

  // ---- sum of exp ----
  float se = 0.0f;
#pragma unroll
  for (int t = 0; t < 4; ++t) se += expf(lv[t] - m);
  for (int off = 16; off > 0; off >>= 1) se += __shfl_xor(se, off, 32);
  if (lane == 0) sred[wave] = se;
  __syncthreads();
  if (wave == 0) {
    float t = sred[lane];
    for (int off = 16; off > 0; off >>= 1) t += __shfl_xor(t, off, 32);
    if (lane == 0) sred[0] = t;
  }
  __syncthreads();
  const float denom = sred[0];

  // ---- init mask to zero ----
#pragma unroll
  for (int t = 0; t < 4; ++t) mask_out[(long)b * N_ + li[t]] = 0.0f;
  __syncthreads();

  // ---- 64 argmax-extraction rounds ----
  for (int r = 0; r < K_; ++r) {
    float v = lv[0]; int idx = li[0];
    argmax_comb(v, idx, lv[1], li[1]);
    argmax_comb(v, idx, lv[2], li[2]);
    argmax_comb(v, idx, lv[3], li[3]);
    for (int off = 16; off > 0; off >>= 1) {
      float ov = __shfl_xor(v, off, 32);
      int   oi = __shfl_xor(idx, off, 32);
      argmax_comb(v, idx, ov, oi);
    }
    if (lane == 0) { wv[wave] = v; wi[wave] = idx; }
    __syncthreads();
    if (wave == 0) {
      float v2 = wv[lane]; int i2 = wi[lane];
      for (int off = 16; off > 0; off >>= 1) {
        float ov = __shfl_xor(v2, off, 32);
        int   oi = __shfl_xor(i2, off, 32);
        argmax_comb(v2, i2, ov, oi);
      }
      if (lane == 0) { sbv = v2; sbi = i2; }
    }
    __syncthreads();
    const int best = sbi;
    const float bval = sbv;
    if ((best >> 2) == tid) lv[best & 3] = -__builtin_huge_valf();  // remove
    if (tid == 0) {
      topk_w[b * K_ + r]  = expf(bval - m) / denom;
      topk_idx[b * K_ + r] = best;
      mask_out[(long)b * N_ + best] = 1.0f;
    }
    __syncthreads();
  }
}

// ---------------------------------------------------------------------------
// Kernel 4: out[b,c] = sum_j topk_w[b,j] * x[b, topk_idx[b,j], c]
// ---------------------------------------------------------------------------
__global__ __launch_bounds__(256)
void wta_gather_kernel(const float* __restrict__ x,
                       const float* __restrict__ topk_w,
                       const int* __restrict__ topk_idx,
                       float* __restrict__ out) {
  const int b = blockIdx.y;
  const int c = blockIdx.x * 256 + threadIdx.x;
  __shared__ float w[K_];
  __shared__ int   idx[K_];
  if (threadIdx.x < K_) {
    w[threadIdx.x]   = topk_w[b * K_ + threadIdx.x];
    idx[threadIdx.x] = topk_idx[b * K_ + threadIdx.x];
  }
  __syncthreads();
  float acc = 0.0f;
#pragma unroll 8
  for (int j = 0; j < K_; ++j)
    acc += w[j] * x[((long)b * N_ + idx[j]) * (long)C_ + c];
  out[b * C_ + c] = acc;
}

// ---------------------------------------------------------------------------
extern "C" void kernel_launch(void* const* d_in, const int* in_sizes, int n_in,
                              void* d_out, int out_size, void* d_ws, size_t ws_size,
                              hipStream_t stream) {
  const float* x = (const float*)d_in[0];   // (B, N, C)
  const float* W = (const float*)d_in[1];   // (C, C)
  const float* b = (const float*)d_in[2];   // (C,)

  float* out  = (float*)d_out;              // (B, C) = 16384 floats
  float* mask = out + B_ * C_;              // (B, N) = 65536 floats

  float* ws       = (float*)d_ws;
  float* w_mean   = ws;                          // 1024
  float* b_mean   = ws + 1024;                   // 1 (padded to 1024)
  float* scores   = ws + 2048;                   // B*N = 65536
  float* topk_w   = scores + (long)B_ * N_;      // B*K = 1024
  int*   topk_idx = (int*)(topk_w + B_ * K_);    // B*K ints

  wta_wmean_kernel<<<dim3(4), dim3(256), 0, stream>>>(W, b, w_mean, b_mean);
  wta_scores_kernel<<<dim3((B_ * N_) / 16 / 8), dim3(256), 0, stream>>>(
      x, w_mean, b_mean, scores);
  wta_topk_kernel<<<dim3(B_), dim3(1024), 0, stream>>>(
      scores, mask, topk_w, topk_idx);
  wta_gather_kernel<<<dim3(C_ / 256, B_), dim3(256), 0, stream>>>(
      x, topk_w, topk_idx, out);
}